// DSGNN_43138651521221
// MI455X (gfx1250) — compile-verified
//
#include <hip/hip_runtime.h>

typedef __attribute__((ext_vector_type(2))) float v2f;
typedef __attribute__((ext_vector_type(8))) float v8f;
typedef __attribute__((ext_vector_type(4))) unsigned int v4u;
typedef __attribute__((ext_vector_type(4))) int v4i;
typedef __attribute__((ext_vector_type(8))) int v8i;

#define TPB 256

// ---------------------------------------------------------------- degree
__global__ void k_init_deg(float* __restrict__ deg, int n) {
  int i = blockIdx.x * blockDim.x + threadIdx.x;
  if (i < n) deg[i] = 1.0f;  // self-loop weight contribution
}

__global__ void k_edge_deg(const int* __restrict__ ei, const float* __restrict__ w,
                           float* __restrict__ deg, int e) {
  int i = blockIdx.x * blockDim.x + threadIdx.x;
  if (i >= e) return;
  float hw = 0.5f * fabsf(w[i]);
  unsafeAtomicAdd(&deg[ei[i]], hw);
  unsafeAtomicAdd(&deg[ei[e + i]], hw);
}

__global__ void k_dinv(float* __restrict__ deg, int n) {  // in place deg -> dinv
  int i = blockIdx.x * blockDim.x + threadIdx.x;
  if (i < n) { float d = deg[i]; deg[i] = (d > 0.0f) ? rsqrtf(d) : 0.0f; }
}

// ------------------------------------------------------- edge coefficients
// forward entry (row=src,col=dst): lr = -a cos(th), li = -a sin(th)
// backward entry (row=dst,col=src): lr same, li negated
__global__ void k_edge_coef(const int* __restrict__ ei, const float* __restrict__ w,
                            const float* __restrict__ dinv,
                            float* __restrict__ c0, float* __restrict__ s0, int e) {
  int i = blockIdx.x * blockDim.x + threadIdx.x;
  if (i >= e) return;
  int s = ei[i], d = ei[e + i];
  float we = w[i];
  float a  = dinv[s] * (0.5f * we) * dinv[d];
  float sn, cs;
  __sincosf(1.5707963267948966f * we, &sn, &cs);  // 2*pi*Q = pi/2
  c0[i] = -a * cs;
  s0[i] = -a * sn;
}

__global__ void k_zero(float* __restrict__ p, int n) {
  int i = blockIdx.x * blockDim.x + threadIdx.x;
  if (i < n) p[i] = 0.0f;
}

// ------------------------------------------ pack W into WMMA B-fragment order
// For each 64x64 matrix: packed[(k16*64 + col)*4 + kk] = W[(k16*4 + kk)*64 + col]
// so a lane's B fragment (K = half*2 + {0,1}, N = col) is one aligned 8-byte pair,
// and a full fragment load covers 256 contiguous LDS bytes (all 64 banks, no movs).
__global__ void k_pack_w(const float* __restrict__ W0, const float* __restrict__ W1,
                         const float* __restrict__ W2, float* __restrict__ Wp,
                         int total) {  // total = 9*4096
  int idx = blockIdx.x * blockDim.x + threadIdx.x;
  if (idx >= total) return;
  int m = idx >> 12;           // matrix 0..8
  int p = idx & 4095;
  int kk  = p & 3;
  int q   = p >> 2;
  int col = q & 63;
  int k16 = q >> 6;
  const float* src = (m < 3) ? W0 : (m < 6) ? W1 : W2;
  src += (size_t)(m % 3) * 4096;
  Wp[idx] = src[(k16 * 4 + kk) * 64 + col];
}

// -------------------------------------------- Chebyshev recurrence init
// out = alpha * diag(i) * a + beta * b   with diag(i) = -dinv(i)^2
__global__ void k_tinit(float* __restrict__ out, const float* __restrict__ a,
                        const float* __restrict__ b, const float* __restrict__ dinv,
                        float alpha, float beta, int nf) {
  int i = blockIdx.x * blockDim.x + threadIdx.x;
  if (i >= nf) return;
  float dv = dinv[i >> 6];
  float v = alpha * (-dv * dv) * a[i];
  if (beta != 0.0f) v += beta * b[i];
  out[i] = v;
}

// -------------------------------------------------- complex edge SpMM
// One wave32 per edge; lane owns features (2*lane, 2*lane+1).
__global__ void k_edge_spmm(const int* __restrict__ ei, const float* __restrict__ c0,
                            const float* __restrict__ s0,
                            const float* __restrict__ ar, const float* __restrict__ ai,
                            float* __restrict__ yr, float* __restrict__ yi,
                            float scale, int e) {
  int gid  = (blockIdx.x * blockDim.x + threadIdx.x) >> 5;
  int lane = threadIdx.x & 31;
  if (gid >= e) return;
  int s = ei[gid], d = ei[e + gid];
  int fo = lane * 2;
  const float* prs = ar + (size_t)s * 64 + fo;
  const float* pis = ai + (size_t)s * 64 + fo;
  const float* prd = ar + (size_t)d * 64 + fo;
  const float* pid = ai + (size_t)d * 64 + fo;
  __builtin_prefetch(prs, 0, 0);
  __builtin_prefetch(pis, 0, 0);
  __builtin_prefetch(prd, 0, 0);
  __builtin_prefetch(pid, 0, 0);
  float c  = c0[gid] * scale;
  float sn = s0[gid] * scale;
  v2f xrs = *(const v2f*)prs;
  v2f xis = *(const v2f*)pis;
  v2f xrd = *(const v2f*)prd;
  v2f xid = *(const v2f*)pid;
  float* oyrd = yr + (size_t)d * 64 + fo;
  float* oyid = yi + (size_t)d * 64 + fo;
  float* oyrs = yr + (size_t)s * 64 + fo;
  float* oyis = yi + (size_t)s * 64 + fo;
  // forward (row=s -> col=d): (lr,li) = (c, sn)
  unsafeAtomicAdd(oyrd + 0, c * xrs.x - sn * xis.x);
  unsafeAtomicAdd(oyrd + 1, c * xrs.y - sn * xis.y);
  unsafeAtomicAdd(oyid + 0, c * xis.x + sn * xrs.x);
  unsafeAtomicAdd(oyid + 1, c * xis.y + sn * xrs.y);
  // backward (row=d -> col=s): (lr,li) = (c, -sn)
  unsafeAtomicAdd(oyrs + 0, c * xrd.x + sn * xid.x);
  unsafeAtomicAdd(oyrs + 1, c * xrd.y + sn * xid.y);
  unsafeAtomicAdd(oyis + 0, c * xid.x - sn * xrd.x);
  unsafeAtomicAdd(oyis + 1, c * xid.y - sn * xrd.y);
}

// --------------------------------------------- WMMA GEMM  Y[n,64] (+)= X[n,64] @ W[64,64]
// 8 waves / block, 16 rows per wave. Wp (packed B-fragment order) is staged
// into LDS via the Tensor Data Mover (1-D D#, 4096 dwords) when available.
// f32 WMMA 16x16x4 layouts (ISA 7.12.2):
//   A: lane(l16,half) holds A[M=l16][K=half*2 + {0,1}] in the 2 VGPRs
//   B: lane(l16,half) holds B[K=half*2 + {0,1}][N=l16]
//   C/D: VGPR j, lane(l16,half) = C[M = j + half*8][N = l16]
__global__ __launch_bounds__(TPB) void k_gemm64(
    const float* __restrict__ Wp, const float* __restrict__ X,
    const float* __restrict__ bias, float* __restrict__ Y,
    int accumulate, int addbias, int n) {
  __shared__ float ldsW[64 * 64];

#if __has_builtin(__builtin_amdgcn_tensor_load_to_lds) && \
    __has_builtin(__builtin_amdgcn_s_wait_tensorcnt)
  if (threadIdx.x < 32) {  // wave 0 issues the DMA (EXEC ignored by TDM)
    unsigned long long ga = (unsigned long long)(uintptr_t)Wp;
    unsigned int lva = (unsigned int)(uintptr_t)ldsW;  // LDS byte address
    v4u g0;
    g0[0] = 1u;                                 // count=1, user descriptor
    g0[1] = lva;                                // lds_addr
    g0[2] = (unsigned int)(ga & 0xFFFFFFFFu);   // global_addr[31:0]
    g0[3] = (unsigned int)((ga >> 32) & 0x1FFFFFFu) | (2u << 30);  // addr[56:32], type=2
    v8i g1;
    g1[0] = 0x20000;              // data_size=2 (4B); no multicast/pad/iterate
    g1[1] = (int)(4096u << 16);   // tensor_dim0 = 4096
    g1[2] = (int)(1u << 16);      // tensor_dim1 = 1
    g1[3] = (int)(4096u << 16);   // tile_dim0 = 4096
    g1[4] = 0;                    // tile_dim1 = 0 (1-D tile), tile_dim2 = 0
    g1[5] = 4096;                 // tensor_dim0_stride
    g1[6] = 0;
    g1[7] = 0;
    v4i gz4 = {0, 0, 0, 0};                   // groups 2/3 unused (<=2D)
    v8i gz8 = {0, 0, 0, 0, 0, 0, 0, 0};       // extended group unused
    __builtin_amdgcn_tensor_load_to_lds(g0, g1, gz4, gz4, gz8, 0);
    __builtin_amdgcn_s_wait_tensorcnt(0);
  }
  __syncthreads();
#else
  {
    const float4* Wv = (const float4*)Wp;
    float4* Lv = (float4*)ldsW;
    for (int i = threadIdx.x; i < 1024; i += TPB) Lv[i] = Wv[i];
  }
  __syncthreads();
#endif

  int wave = threadIdx.x >> 5;
  int lane = threadIdx.x & 31;
  int half = lane >> 4;
  int l16  = lane & 15;
  int m0   = blockIdx.x * 128 + wave * 16;
  if (m0 >= n) return;  // wave-uniform: EXEC stays all-ones for WMMA

  const float* xrow = X + (size_t)(m0 + l16) * 64;

  v8f acc[4];
  if (accumulate) {
#pragma unroll
    for (int nt = 0; nt < 4; ++nt)
#pragma unroll
      for (int j = 0; j < 8; ++j)
        acc[nt][j] = Y[(size_t)(m0 + half * 8 + j) * 64 + nt * 16 + l16];
  } else {
#pragma unroll
    for (int nt = 0; nt < 4; ++nt) acc[nt] = (v8f)0.0f;
  }

#pragma unroll
  for (int k16 = 0; k16 < 16; ++k16) {
    v2f a;
    a.x = xrow[k16 * 4 + half * 2];
    a.y = xrow[k16 * 4 + half * 2 + 1];
#pragma unroll
    for (int nt = 0; nt < 4; ++nt) {
      // packed: one aligned 8-byte pair per lane, 256B contiguous per fragment
      v2f b = *(const v2f*)(ldsW + ((k16 * 64 + nt * 16 + l16) << 2) + half * 2);
      acc[nt] = __builtin_amdgcn_wmma_f32_16x16x4_f32(
          false, a, false, b, (short)0, acc[nt], false, false);
    }
  }

#pragma unroll
  for (int nt = 0; nt < 4; ++nt) {
    float bv = addbias ? bias[nt * 16 + l16] : 0.0f;
#pragma unroll
    for (int j = 0; j < 8; ++j)
      Y[(size_t)(m0 + half * 8 + j) * 64 + nt * 16 + l16] = acc[nt][j] + bv;
  }
}

// ------------------------------------------------------------- pooling
__global__ void k_pool(const float* __restrict__ outr, const float* __restrict__ outi,
                       const int* __restrict__ batch, float* __restrict__ pooled,
                       int offr, int offi, int nf) {
  int i = blockIdx.x * blockDim.x + threadIdx.x;
  if (i >= nf) return;
  int node = i >> 6, f = i & 63;
  int g = batch[node];
  unsafeAtomicAdd(pooled + g * 384 + offr + f, outr[i]);
  unsafeAtomicAdd(pooled + g * 384 + offi + f, outi[i]);
}

// complex ReLU (mask by sign of real part), in place
__global__ void k_relu(float* __restrict__ r, float* __restrict__ im, int nf) {
  int i = blockIdx.x * blockDim.x + threadIdx.x;
  if (i >= nf) return;
  if (r[i] < 0.0f) { r[i] = 0.0f; im[i] = 0.0f; }
}

// final  [64,384] @ [384,32] + bl
__global__ void k_final(const float* __restrict__ pooled, const float* __restrict__ Wl,
                        const float* __restrict__ bl, float* __restrict__ out) {
  int idx = blockIdx.x * blockDim.x + threadIdx.x;
  if (idx >= 64 * 32) return;
  int g = idx >> 5, o = idx & 31;
  float acc = bl[o];
#pragma unroll 8
  for (int f = 0; f < 384; ++f) acc += pooled[g * 384 + f] * Wl[f * 32 + o];
  out[idx] = acc;
}

// ================================================================= host
extern "C" void kernel_launch(void* const* d_in, const int* in_sizes, int n_in,
                              void* d_out, int out_size, void* d_ws, size_t ws_size,
                              hipStream_t stream) {
  const float* x  = (const float*)d_in[0];
  const float* ew = (const float*)d_in[1];
  const float* Wl = (const float*)d_in[8];
  const float* bl = (const float*)d_in[9];
  const int* ei    = (const int*)d_in[10];
  const int* batch = (const int*)d_in[11];
  float* out = (float*)d_out;

  const int N  = in_sizes[0] / 64;   // 65536
  const int E  = in_sizes[1];        // 1048576
  const int NF = N * 64;

  // ---- workspace carve-up (floats) ----
  float* pw     = (float*)d_ws;
  float* dinv   = pw;                 pw += N;
  float* c0     = pw;                 pw += E;
  float* s0     = pw;                 pw += E;
  float* pooled = pw;                 pw += 64 * 384;
  float* Wpck   = pw;                 pw += 9 * 4096;  // 9 packed 64x64 weights
  float* T1R = pw;  pw += NF;
  float* T1I = pw;  pw += NF;
  float* T2R = pw;  pw += NF;
  float* T2I = pw;  pw += NF;
  float* OAR = pw;  pw += NF;
  float* OAI = pw;  pw += NF;
  float* OBR = pw;  pw += NF;
  float* OBI = pw;  pw += NF;

  const int gN  = (N + TPB - 1) / TPB;
  const int gE  = (E + TPB - 1) / TPB;
  const int gNF = (NF + TPB - 1) / TPB;
  const int gEW = (E * 32 + TPB - 1) / TPB;   // one wave per edge
  const int gGM = (N + 127) / 128;            // gemm: 128 rows / block

  // ---- Laplacian prep + weight packing ----
  k_init_deg<<<gN, TPB, 0, stream>>>(dinv, N);
  k_edge_deg<<<gE, TPB, 0, stream>>>(ei, ew, dinv, E);
  k_dinv<<<gN, TPB, 0, stream>>>(dinv, N);
  k_edge_coef<<<gE, TPB, 0, stream>>>(ei, ew, dinv, c0, s0, E);
  k_zero<<<(64 * 384 + TPB - 1) / TPB, TPB, 0, stream>>>(pooled, 64 * 384);
  k_pack_w<<<(9 * 4096 + TPB - 1) / TPB, TPB, 0, stream>>>(
      (const float*)d_in[2], (const float*)d_in[4], (const float*)d_in[6],
      Wpck, 9 * 4096);

  // ---- 3 ChebConv layers ----
  const float* Xr = x;
  const float* Xi = x;  // hr = hi = x initially
  for (int l = 0; l < 3; ++l) {
    const float* b = (const float*)d_in[3 + 2 * l];   // [64]
    const float* Wk0 = Wpck + (size_t)(l * 3 + 0) * 4096;
    const float* Wk1 = Wpck + (size_t)(l * 3 + 1) * 4096;
    const float* Wk2 = Wpck + (size_t)(l * 3 + 2) * 4096;
    float* OR = (l == 1) ? OBR : OAR;
    float* OI = (l == 1) ? OBI : OAI;

    // out = T0 @ W[0]
    k_gemm64<<<gGM, TPB, 0, stream>>>(Wk0, Xr, b, OR, 0, 0, N);
    k_gemm64<<<gGM, TPB, 0, stream>>>(Wk0, Xi, b, OI, 0, 0, N);

    // T1 = L_hat X   (diag part in init, edges via atomics)
    k_tinit<<<gNF, TPB, 0, stream>>>(T1R, Xr, Xr, dinv, 1.0f, 0.0f, NF);
    k_tinit<<<gNF, TPB, 0, stream>>>(T1I, Xi, Xi, dinv, 1.0f, 0.0f, NF);
    k_edge_spmm<<<gEW, TPB, 0, stream>>>(ei, c0, s0, Xr, Xi, T1R, T1I, 1.0f, E);

    // out += T1 @ W[1]
    k_gemm64<<<gGM, TPB, 0, stream>>>(Wk1, T1R, b, OR, 1, 0, N);
    k_gemm64<<<gGM, TPB, 0, stream>>>(Wk1, T1I, b, OI, 1, 0, N);

    // T2 = 2 L_hat T1 - T0
    k_tinit<<<gNF, TPB, 0, stream>>>(T2R, T1R, Xr, dinv, 2.0f, -1.0f, NF);
    k_tinit<<<gNF, TPB, 0, stream>>>(T2I, T1I, Xi, dinv, 2.0f, -1.0f, NF);
    k_edge_spmm<<<gEW, TPB, 0, stream>>>(ei, c0, s0, T1R, T1I, T2R, T2I, 2.0f, E);

    // out += T2 @ W[2] + bias
    k_gemm64<<<gGM, TPB, 0, stream>>>(Wk2, T2R, b, OR, 1, 1, N);
    k_gemm64<<<gGM, TPB, 0, stream>>>(Wk2, T2I, b, OI, 1, 1, N);

    // pool pre-activation hiddens, then ReLU in place for next layer
    k_pool<<<gNF, TPB, 0, stream>>>(OR, OI, batch, pooled, 64 * l, 192 + 64 * l, NF);
    k_relu<<<gNF, TPB, 0, stream>>>(OR, OI, NF);

    Xr = OR;
    Xi = OI;
  }

  // ---- readout ----
  k_final<<<(64 * 32 + TPB - 1) / TPB, TPB, 0, stream>>>(pooled, Wl, bl, out);
}